// SKIPGAT_11948599017910
// MI455X (gfx1250) — compile-verified
//
#include <hip/hip_runtime.h>
#include <hip/hip_bf16.h>

#define NNODES 50000
#define NEDGES 800000
#define FDIM   256      // H*C = 8*32, also input feature dim
#define NHEAD  8
#define CHEAD  32
#define EDIM   16
#define LDA    260      // padded LDS row stride (floats): conflict-free for WMMA A fetch

typedef __attribute__((ext_vector_type(2))) float v2f;
typedef __attribute__((ext_vector_type(8))) float v8f;

// ---------------------------------------------------------------------------
// w_e[d][h] = sum_c W_e[d, h*32+c] * att_e[h*32+c]   (ED x H = 16 x 8)
// ---------------------------------------------------------------------------
__global__ void gat_prep_we(const float* __restrict__ W_e,
                            const float* __restrict__ att_e,
                            float* __restrict__ w_e) {
    int idx = threadIdx.x;               // 128 threads: d = idx>>3, h = idx&7
    if (idx >= EDIM * NHEAD) return;
    int d = idx >> 3, h = idx & 7;
    float s = 0.f;
#pragma unroll
    for (int c = 0; c < CHEAD; ++c)
        s += W_e[d * FDIM + h * CHEAD + c] * att_e[h * CHEAD + c];
    w_e[d * NHEAD + h] = s;
}

// ---------------------------------------------------------------------------
// C[N x 256] = relu(A[N x 256]) @ B[256 x 256]  using V_WMMA_F32_16X16X4_F32.
// Block: 512 threads = 16 waves; block owns a 16-row stripe of A (in LDS),
// each wave owns one 16-col tile of C. grid.x = N/16 = 3125.
// ---------------------------------------------------------------------------
__global__ __launch_bounds__(512)
void gat_gemm_relu_wmma(const float* __restrict__ A,
                        const float* __restrict__ B,
                        float* __restrict__ C,
                        int apply_relu) {
    __shared__ float As[16 * LDA];
    const int m0 = blockIdx.x * 16;

    // Stage relu(A) stripe: 16 rows x 256 cols as float4 (1024 vec4 / 512 thr)
    for (int i = threadIdx.x; i < 16 * (FDIM / 4); i += 512) {
        int r  = i >> 6;            // 64 float4 per row
        int c4 = (i & 63) << 2;
        float4 v = *(const float4*)(A + (size_t)(m0 + r) * FDIM + c4);
        if (apply_relu) {
            v.x = fmaxf(v.x, 0.f); v.y = fmaxf(v.y, 0.f);
            v.z = fmaxf(v.z, 0.f); v.w = fmaxf(v.w, 0.f);
        }
        float* p = &As[r * LDA + c4];
        p[0] = v.x; p[1] = v.y; p[2] = v.z; p[3] = v.w;
    }
    __syncthreads();

    const int lane = threadIdx.x & 31;
    const int wv   = threadIdx.x >> 5;
    const int n0   = wv * 16;             // this wave's 16-col tile
    const int mrow = lane & 15;           // M (for A) / N (for B) index
    const int koff = (lane >> 4) * 2;     // lanes 16-31 hold K+2,K+3

    v8f acc = {};                         // 16x16 f32 accumulator (8 VGPRs)
#pragma unroll 4
    for (int k0 = 0; k0 < FDIM; k0 += 4) {
        v2f a, b;
        // A 16x4 tile: lane holds A[mrow][k0+koff], A[mrow][k0+koff+1]
        a.x = As[mrow * LDA + k0 + koff];
        a.y = As[mrow * LDA + k0 + koff + 1];
        // B 4x16 tile: lane holds B[k0+koff][n0+mrow], B[k0+koff+1][n0+mrow]
        const float* Bp = B + (size_t)(k0 + koff) * FDIM + n0 + mrow;
        b.x = Bp[0];
        b.y = Bp[FDIM];
        acc = __builtin_amdgcn_wmma_f32_16x16x4_f32(
                  false, a, false, b, (short)0, acc, false, false);
    }

    // C/D layout: VGPR i -> row = i + 8*(lane>>4), col = n0 + (lane&15)
    const int colg  = n0 + mrow;
    const int rbase = m0 + (lane >> 4) * 8;
#pragma unroll
    for (int i = 0; i < 8; ++i)
        C[(size_t)(rbase + i) * FDIM + colg] = acc[i];
}

// ---------------------------------------------------------------------------
// a_src[n,h] = xp[n,h,:].att_src[h,:],  a_dst likewise. One thread per (n,h).
// ---------------------------------------------------------------------------
__global__ void gat_att_dots(const float* __restrict__ xp,
                             const float* __restrict__ att_src,
                             const float* __restrict__ att_dst,
                             float* __restrict__ a_src,
                             float* __restrict__ a_dst) {
    int idx = blockIdx.x * blockDim.x + threadIdx.x;
    if (idx >= NNODES * NHEAD) return;
    int h = idx & 7, n = idx >> 3;
    const float* xr = xp + (size_t)n * FDIM + h * CHEAD;
    const float* as = att_src + h * CHEAD;
    const float* ad = att_dst + h * CHEAD;
    float ss = 0.f, sd = 0.f;
#pragma unroll
    for (int c = 0; c < CHEAD; ++c) {
        float v = xr[c];
        ss += v * as[c];
        sd += v * ad[c];
    }
    a_src[idx] = ss;
    a_dst[idx] = sd;
}

// ---------------------------------------------------------------------------
// init: msg=0 over N*256; m=-inf, denom=0 over N*8
// ---------------------------------------------------------------------------
__global__ void gat_init(float* __restrict__ mbuf,
                         float* __restrict__ denom,
                         float* __restrict__ msg) {
    int idx = blockIdx.x * blockDim.x + threadIdx.x;
    if (idx < NNODES * FDIM) msg[idx] = 0.f;
    if (idx < NNODES * NHEAD) {
        mbuf[idx]  = -__builtin_inff();
        denom[idx] = 0.f;
    }
}

__device__ __forceinline__ void atomicMaxFloat(float* addr, float val) {
    // Sign-split trick: valid with addr initialized to -inf.
    if (val >= 0.f)
        atomicMax((int*)addr, __float_as_int(val));
    else
        atomicMin((unsigned int*)addr, __float_as_uint(val));
}

// ---------------------------------------------------------------------------
// Pass 1: alpha[e,h] = leaky_relu(a_src[src]+a_dst[dst]+edge_attr.w_e[:,h]);
//         segment max into mbuf[dst,h]. One thread per (e,h).
// ---------------------------------------------------------------------------
__global__ void gat_edge_alpha(const int* __restrict__ src,
                               const int* __restrict__ dst,
                               const float* __restrict__ edge_attr,
                               const float* __restrict__ a_src,
                               const float* __restrict__ a_dst,
                               const float* __restrict__ w_e,
                               float* __restrict__ alpha,
                               float* __restrict__ mbuf) {
    int idx = blockIdx.x * blockDim.x + threadIdx.x;
    if (idx >= NEDGES * NHEAD) return;
    int h = idx & 7, e = idx >> 3;
    int s = src[e], d = dst[e];
    const float* ea = edge_attr + (size_t)e * EDIM;
    float aedge = 0.f;
#pragma unroll
    for (int dd = 0; dd < EDIM; ++dd)
        aedge += ea[dd] * w_e[dd * NHEAD + h];
    float a = a_src[s * NHEAD + h] + a_dst[d * NHEAD + h] + aedge;
    a = (a > 0.f) ? a : 0.2f * a;     // leaky relu
    alpha[idx] = a;
    atomicMaxFloat(&mbuf[d * NHEAD + h], a);
}

// ---------------------------------------------------------------------------
// Pass 2: ex = exp(alpha - m[dst]);  msg[dst,hc] += ex * xp[src,hc];
//         denom[dst,h] += ex (lane c==0). One thread per (e, hc). Softmax
// normalization is factored out of the edge loop (done per-node later).
// ---------------------------------------------------------------------------
__global__ void gat_edge_scatter(const int* __restrict__ src,
                                 const int* __restrict__ dst,
                                 const float* __restrict__ alpha,
                                 const float* __restrict__ mbuf,
                                 const float* __restrict__ xp,
                                 float* __restrict__ msg,
                                 float* __restrict__ denom) {
    int idx = blockIdx.x * blockDim.x + threadIdx.x;   // < E*256 = 204.8M
    int e  = idx >> 8;
    int hc = idx & 255;
    int h  = hc >> 5;
    int s = src[e], d = dst[e];
    float ex = __expf(alpha[e * NHEAD + h] - mbuf[d * NHEAD + h]);
    float v  = xp[(size_t)s * FDIM + hc];
    atomicAdd(&msg[(size_t)d * FDIM + hc], ex * v);
    if ((hc & 31) == 0)
        atomicAdd(&denom[d * NHEAD + h], ex);
}

// ---------------------------------------------------------------------------
// out[n,hc] = msg[n,hc] / (denom[n,h] + 1e-16) + bias[hc] (+ skip[n,hc])
// ---------------------------------------------------------------------------
__global__ void gat_normalize(const float* __restrict__ msg,
                              const float* __restrict__ denom,
                              const float* __restrict__ bias,
                              const float* __restrict__ skip,
                              float* __restrict__ out) {
    int idx = blockIdx.x * blockDim.x + threadIdx.x;
    if (idx >= NNODES * FDIM) return;
    int hc = idx & 255;
    int h  = hc >> 5;
    int n  = idx >> 8;
    float v = msg[idx] / (denom[n * NHEAD + h] + 1e-16f) + bias[hc];
    if (skip) v += skip[idx];
    out[idx] = v;
}

// ---------------------------------------------------------------------------
extern "C" void kernel_launch(void* const* d_in, const int* in_sizes, int n_in,
                              void* d_out, int out_size, void* d_ws, size_t ws_size,
                              hipStream_t stream) {
    const float* x        = (const float*)d_in[0];
    const int*   ei       = (const int*)  d_in[1];
    const float* eattr    = (const float*)d_in[2];
    const float* W1       = (const float*)d_in[3];
    const float* att_src1 = (const float*)d_in[4];
    const float* att_dst1 = (const float*)d_in[5];
    const float* W_e1     = (const float*)d_in[6];
    const float* att_e1   = (const float*)d_in[7];
    const float* b1       = (const float*)d_in[8];
    const float* W2       = (const float*)d_in[9];
    const float* att_src2 = (const float*)d_in[10];
    const float* att_dst2 = (const float*)d_in[11];
    const float* W_e2     = (const float*)d_in[12];
    const float* att_e2   = (const float*)d_in[13];
    const float* b2       = (const float*)d_in[14];

    const int* src = ei;            // edge_index[0,:]
    const int* dst = ei + NEDGES;   // edge_index[1,:]

    float* ws    = (float*)d_ws;
    float* xp    = ws;                              // N*256
    float* msg   = xp    + (size_t)NNODES * FDIM;   // N*256
    float* alpha = msg   + (size_t)NNODES * FDIM;   // E*8
    float* a_src = alpha + (size_t)NEDGES * NHEAD;  // N*8
    float* a_dst = a_src + (size_t)NNODES * NHEAD;  // N*8
    float* mbuf  = a_dst + (size_t)NNODES * NHEAD;  // N*8
    float* denom = mbuf  + (size_t)NNODES * NHEAD;  // N*8
    float* w_e   = denom + (size_t)NNODES * NHEAD;  // 16*8
    float* out   = (float*)d_out;                   // also h1 temp

    const int gGemm  = NNODES / 16;                         // 3125
    const int gNH    = (NNODES * NHEAD + 255) / 256;        // a_src/a_dst
    const int gNF    = (NNODES * FDIM) / 256;               // 50000
    const int gEH    = (NEDGES * NHEAD) / 256;              // 25000
    const int gEF    = NEDGES;                              // E*256/256

    // ---------------- conv1: h1 = GAT(relu(x)) ----------------
    gat_prep_we      <<<1, 128, 0, stream>>>(W_e1, att_e1, w_e);
    gat_gemm_relu_wmma<<<gGemm, 512, 0, stream>>>(x, W1, xp, 1);
    gat_att_dots     <<<gNH, 256, 0, stream>>>(xp, att_src1, att_dst1, a_src, a_dst);
    gat_init         <<<gNF, 256, 0, stream>>>(mbuf, denom, msg);
    gat_edge_alpha   <<<gEH, 256, 0, stream>>>(src, dst, eattr, a_src, a_dst, w_e, alpha, mbuf);
    gat_edge_scatter <<<gEF, 256, 0, stream>>>(src, dst, alpha, mbuf, xp, msg, denom);
    gat_normalize    <<<gNF, 256, 0, stream>>>(msg, denom, b1, nullptr, out);   // out = h1

    // ---------------- conv2: out = GAT(relu(h1)) + x ----------------
    gat_prep_we      <<<1, 128, 0, stream>>>(W_e2, att_e2, w_e);
    gat_gemm_relu_wmma<<<gGemm, 512, 0, stream>>>(out, W2, xp, 1);
    gat_att_dots     <<<gNH, 256, 0, stream>>>(xp, att_src2, att_dst2, a_src, a_dst);
    gat_init         <<<gNF, 256, 0, stream>>>(mbuf, denom, msg);
    gat_edge_alpha   <<<gEH, 256, 0, stream>>>(src, dst, eattr, a_src, a_dst, w_e, alpha, mbuf);
    gat_edge_scatter <<<gEF, 256, 0, stream>>>(src, dst, alpha, mbuf, xp, msg, denom);
    gat_normalize    <<<gNF, 256, 0, stream>>>(msg, denom, b2, x, out);         // + skip
}